// PartialChamferLoss_42795054137807
// MI455X (gfx1250) — compile-verified
//
#include <hip/hip_runtime.h>
#include <cstdint>
#include <cstddef>

// Problem constants from the reference: B=2, N=8192, M=32768, S=4096
#define B_  2
#define N_  8192
#define M_  32768
#define S_  4096
#define TILES_PER_B   (M_ / 16)          // 2048 target tiles per batch
#define CHUNKS        8                  // split M across 8 waves per s-tile
#define TILES_PER_CHK (TILES_PER_B / CHUNKS) // 256
#define STILES        (B_ * S_ / 16)     // 512 s-tiles total

typedef __attribute__((ext_vector_type(2))) float v2f;
typedef __attribute__((ext_vector_type(8))) float v8f;

// ---------------------------------------------------------------------------
// Kernel 1: pre-pack target data into per-lane WMMA B-matrix tiles.
// B is 4x16 (KxN) f32, 2 VGPRs/lane:
//   lanes 0-15 : VGPR0 = B[K=0][n]= -2*tx, VGPR1 = B[K=1][n] = -2*ty   (n = lane)
//   lanes 16-31: VGPR0 = B[K=2][n]= -2*tz, VGPR1 = B[K=3][n] = y2     (n = lane-16)
// Stored as btiles[tile][lane] : float2  (one b64 load per lane per tile).
// ---------------------------------------------------------------------------
__global__ void __launch_bounds__(256)
chamfer_pack_b(const float* __restrict__ target, float* __restrict__ btiles) {
    int g = blockIdx.x * blockDim.x + threadIdx.x;      // 0 .. B*M-1
    if (g >= B_ * M_) return;
    int b = g / M_;
    int m = g - b * M_;
    float tx = target[(size_t)g * 3 + 0];
    float ty = target[(size_t)g * 3 + 1];
    float tz = target[(size_t)g * 3 + 2];
    float y2 = tx * tx + ty * ty + tz * tz;
    int   tg = b * TILES_PER_B + (m >> 4);
    int   l  = m & 15;
    float2* out = (float2*)btiles;
    out[(size_t)tg * 32 + l]      = make_float2(-2.0f * tx, -2.0f * ty);
    out[(size_t)tg * 32 + 16 + l] = make_float2(-2.0f * tz, y2);
}

// ---------------------------------------------------------------------------
// Kernel 2: main WMMA loop. One wave handles one (s-tile, M-chunk):
// builds A (16x4: [px,py,pz,1]) and C (x2 per row) once, then loops over
// 256 target tiles: load b64 -> v_wmma_f32_16x16x4_f32 -> 8x v_min_i32.
//
// Integer-min trick: for non-negative f32, signed-i32 compare == f32 compare.
// Cancellation negatives (d2 slightly < 0) all win the signed-int min over
// any positive, and every negative maps to distance 0 via the final
// fmaxf(mn,0) clamp — identical to the reference's sqrt(max(d2,0)) + min.
// Integer mins need no FP canonicalization -> no v_max x,x noise.
// ---------------------------------------------------------------------------
__global__ void __launch_bounds__(256)
chamfer_wmma(const float* __restrict__ pred,
             const long long* __restrict__ idx,
             const float* __restrict__ btiles,
             float* __restrict__ partial) {
    const int lane  = threadIdx.x & 31;
    const int wave  = threadIdx.x >> 5;
    const int wgid  = blockIdx.x * (blockDim.x >> 5) + wave; // 0 .. STILES*CHUNKS-1
    const int st    = wgid >> 3;          // s-tile 0..511
    const int chunk = wgid & (CHUNKS - 1);
    const int b     = st >> 8;            // 256 s-tiles per batch
    const int s0    = (st & 255) << 4;    // s offset within batch

    // --- A matrix (16x4 f32): lane r & r+16 both handle row r = lane&15 ---
    const int  row = lane & 15;
    const long long j = idx[s0 + row];    // gather index into pred N-dim
    const float* pp = pred + ((size_t)b * N_ + (size_t)j) * 3;
    float px = pp[0], py = pp[1], pz = pp[2];
    float x2 = px * px + py * py + pz * pz;   // identical in paired lanes

    v2f a;
    if (lane < 16) { a.x = px; a.y = py; }    // K0,K1
    else           { a.x = pz; a.y = 1.0f; }  // K2,K3

    // --- C matrix: c[i] = x2 of row (i + 8*(lane>=16)); x2[row r] lives in lane r ---
    v8f cinit;
    const int rbase = (lane & 16) ? 8 : 0;
#pragma unroll
    for (int i = 0; i < 8; ++i)
        cinit[i] = __shfl(x2, rbase + i);

    // --- hot loop over this chunk's target tiles ---
    const float2* bt = (const float2*)btiles
                     + ((size_t)(b * TILES_PER_B + chunk * TILES_PER_CHK) * 32 + lane);
    int minv[8];
#pragma unroll
    for (int i = 0; i < 8; ++i) minv[i] = 0x7F7FFFFF;   // bits of FLT_MAX

#pragma unroll 8
    for (int t = 0; t < TILES_PER_CHK; ++t) {
        float2 bl = bt[(size_t)t * 32];
        v2f bv; bv.x = bl.x; bv.y = bl.y;
        // D = A*B + C  ->  d2[s][m] tile (16x16 f32)
        v8f d = __builtin_amdgcn_wmma_f32_16x16x4_f32(
            /*neg_a=*/false, a, /*neg_b=*/false, bv,
            /*c_mod=*/(short)0, cinit, /*reuse_a=*/false, /*reuse_b=*/false);
#pragma unroll
        for (int i = 0; i < 8; ++i)
            minv[i] = min(minv[i], __float_as_int(d[i]));   // v_min_i32
    }

    // --- min across the 16 columns (lanes within each half) ---
#pragma unroll
    for (int off = 8; off >= 1; off >>= 1) {
#pragma unroll
        for (int i = 0; i < 8; ++i)
            minv[i] = min(minv[i], __shfl_xor(minv[i], off));
    }

    // lane 0 holds rows 0..7, lane 16 holds rows 8..15
    if ((lane & 15) == 0) {
#pragma unroll
        for (int i = 0; i < 8; ++i) {
            size_t s_glob = (size_t)b * S_ + s0 + rbase + i;
            partial[s_glob * CHUNKS + chunk] = __int_as_float(minv[i]);
        }
    }
}

// ---------------------------------------------------------------------------
// Kernel 3: deterministic final reduction: min over chunks, clamp+sqrt, mean.
// Single block; plain '=' store into d_out (harness poisons d_out).
// ---------------------------------------------------------------------------
__global__ void __launch_bounds__(256)
chamfer_reduce(const float* __restrict__ partial, float* __restrict__ out) {
    __shared__ float sm[256];
    const int t = threadIdx.x;
    float acc = 0.0f;
    for (int s = t; s < B_ * S_; s += 256) {
        // same signed-int min: partials are d2 values (possibly tiny negatives)
        int mn = __float_as_int(partial[(size_t)s * CHUNKS + 0]);
#pragma unroll
        for (int c = 1; c < CHUNKS; ++c)
            mn = min(mn, __float_as_int(partial[(size_t)s * CHUNKS + c]));
        acc += sqrtf(fmaxf(__int_as_float(mn), 0.0f));  // negatives -> 0
    }
    sm[t] = acc;
    __syncthreads();
    for (int off = 128; off > 0; off >>= 1) {
        if (t < off) sm[t] += sm[t + off];
        __syncthreads();
    }
    if (t == 0) out[0] = sm[0] * (1.0f / (float)(B_ * S_));
}

// ---------------------------------------------------------------------------
extern "C" void kernel_launch(void* const* d_in, const int* in_sizes, int n_in,
                              void* d_out, int out_size, void* d_ws, size_t ws_size,
                              hipStream_t stream) {
    (void)in_sizes; (void)n_in; (void)out_size; (void)ws_size;
    const float*     pred   = (const float*)d_in[0];
    const float*     target = (const float*)d_in[1];
    const long long* idx    = (const long long*)d_in[2]; // reference dtype: int64
    float*           out    = (float*)d_out;

    // workspace layout (floats): [partial: B*S*CHUNKS][btiles: B*TILES_PER_B*32*2]
    float* partial = (float*)d_ws;                       // 65536 floats (256 KB)
    float* btiles  = partial + (size_t)B_ * S_ * CHUNKS; // 262144 floats (1 MB)

    chamfer_pack_b<<<(B_ * M_ + 255) / 256, 256, 0, stream>>>(target, btiles);
    // 4096 waves, 8 waves/block -> 512 blocks
    chamfer_wmma<<<(STILES * CHUNKS) / 8, 256, 0, stream>>>(pred, idx, btiles, partial);
    chamfer_reduce<<<1, 256, 0, stream>>>(partial, out);
}